// ComplexGaussianTracer_25151328485676
// MI455X (gfx1250) — compile-verified
//
#include <hip/hip_runtime.h>
#include <math.h>

#define H_DIM 128
#define W_DIM 128
#define NPIX (H_DIM * W_DIM)
#define NG 2048

typedef __attribute__((ext_vector_type(2))) float v2f;
typedef __attribute__((ext_vector_type(8))) float v8f;
typedef __attribute__((ext_vector_type(4))) unsigned u32x4;
typedef __attribute__((ext_vector_type(4))) int i32x4;
typedef __attribute__((ext_vector_type(8))) int i32x8;

#if defined(__has_builtin)
#if __has_builtin(__builtin_amdgcn_tensor_load_to_lds) && \
    __has_builtin(__builtin_amdgcn_s_wait_tensorcnt)
#define USE_TDM 1
#endif
#endif
#ifndef USE_TDM
#define USE_TDM 0
#endif

__device__ __forceinline__ float wrap_pi(float x) {
  // (x + pi) mod 2pi - pi, floor-mod (JAX semantics)
  const float TWO_PI = 6.28318530717958647692f;
  const float PI_F   = 3.14159265358979323846f;
  float t = x + PI_F;
  t = t - TWO_PI * floorf(t * (1.0f / TWO_PI));
  return t - PI_F;
}

// ---------------------------------------------------------------------------
// Kernel 1: per-Gaussian preprocessing.
// Emits {theta, phi, -ia/2, b/det, -ic/2, att, s0, s1} so the raster loop
// computes the exp() argument directly: e = p2*dt^2 + p3*dt*dp + p4*dp^2.
// ---------------------------------------------------------------------------
__global__ void preprocess_kernel(const float* __restrict__ means,
                                  const float* __restrict__ cov6,
                                  const float* __restrict__ sig,
                                  const float* __restrict__ atten,
                                  const float* __restrict__ radii,
                                  const float* __restrict__ rx,
                                  float* __restrict__ dist_out,
                                  float* __restrict__ params_out) {
  int i = blockIdx.x * blockDim.x + threadIdx.x;
  if (i >= NG) return;

  const float PI_F = 3.14159265358979323846f;

  float dx = means[i * 3 + 0] - rx[0];
  float dy = means[i * 3 + 1] - rx[1];
  float dz = means[i * 3 + 2] - rx[2];
  float d2 = dx * dx + dy * dy + dz * dz + 1e-12f;
  float dist = sqrtf(d2);
  float keep = (dist > 0.75f) ? 1.0f : 0.0f;  // SCALE_DIS * RADIUS_RX
  float inv = 1.0f / dist;
  float ux = dx * inv, uy = dy * inv, uz = dz * inv;

  float st2 = fmaxf(ux * ux + uy * uy, 1e-12f);
  float sin_t = sqrtf(st2);
  float cos_t = uz;
  float theta = acosf(fminf(fmaxf(uz, -1.0f), 1.0f));
  float phi = atan2f(uy, ux);
  float isin = 1.0f / sin_t;
  float cos_p = ux * isin, sin_p = uy * isin;

  float et0 = cos_t * cos_p, et1 = cos_t * sin_p, et2 = -sin_t;
  float ep0 = -sin_p,        ep1 = cos_p,         ep2 = 0.0f;

  float c0 = cov6[i * 6 + 0], c1 = cov6[i * 6 + 1], c2 = cov6[i * 6 + 2];
  float c3 = cov6[i * 6 + 3], c4 = cov6[i * 6 + 4], c5 = cov6[i * 6 + 5];

  float vt0 = c0 * et0 + c1 * et1 + c2 * et2;
  float vt1 = c1 * et0 + c3 * et1 + c4 * et2;
  float vt2 = c2 * et0 + c4 * et1 + c5 * et2;
  float vp0 = c0 * ep0 + c1 * ep1 + c2 * ep2;
  float vp1 = c1 * ep0 + c3 * ep1 + c4 * ep2;
  float vp2 = c2 * ep0 + c4 * ep1 + c5 * ep2;

  float invd2 = 1.0f / d2;
  float cov00 = (et0 * vt0 + et1 * vt1 + et2 * vt2) * invd2;
  float cov01 = (et0 * vp0 + et1 * vp1 + et2 * vp2) * invd2;
  float cov11 = (ep0 * vp0 + ep1 * vp1 + ep2 * vp2) * invd2;

  float pix_blur = 0.3f * (PI_F / H_DIM) * (PI_F / H_DIM);
  float gr = radii[i] * inv;
  float iso = gr * gr + pix_blur;

  float a = cov00 + iso, b = cov01, cc = cov11 + iso;
  float det = fmaxf(a * cc - b * b, 1e-12f);
  float idet = 1.0f / det;

  dist_out[i] = dist;
  float* p = params_out + i * 8;
  p[0] = theta;
  p[1] = phi;
  p[2] = -0.5f * cc * idet;  // -ia/2
  p[3] = b * idet;           // -(2*ib)/2 = +b/det
  p[4] = -0.5f * a * idet;   // -ic/2
  p[5] = atten[i] * keep;
  p[6] = sig[i * 2 + 0];
  p[7] = sig[i * 2 + 1];
}

// ---------------------------------------------------------------------------
// Kernel 2: O(N^2) stable rank sort by distance + scatter (N=2048, trivial).
// ---------------------------------------------------------------------------
__global__ void sort_scatter_kernel(const float* __restrict__ dist,
                                    const float* __restrict__ params_in,
                                    float* __restrict__ params_sorted) {
  __shared__ float sdist[NG];
  for (int k = threadIdx.x; k < NG; k += blockDim.x) sdist[k] = dist[k];
  __syncthreads();

  int i = blockIdx.x * blockDim.x + threadIdx.x;
  if (i >= NG) return;
  float di = sdist[i];
  int rank = 0;
  for (int j = 0; j < NG; ++j) {
    float dj = sdist[j];
    rank += (dj < di) || ((dj == di) && (j < i));
  }
  const float4* src = (const float4*)(params_in + (size_t)i * 8);
  float4* dst = (float4*)(params_sorted + (size_t)rank * 8);
  dst[0] = src[0];
  dst[1] = src[1];
}

// ---------------------------------------------------------------------------
// Kernel 3: rasterize. 256 threads = 8 waves; each wave owns 16 pixels.
// half = lane>>4 picks Gaussian pair {2h,2h+1} of each group of 4;
// col = lane&15 is the pixel row (WMMA M). Weights land directly in the
// V_WMMA_F32_16X16X4_F32 A layout; B carries sig; 16x16 f32 C/D = image.
// Gaussian table staged into LDS by the Tensor Data Mover (one 64KB DMA).
// ---------------------------------------------------------------------------
__global__ __launch_bounds__(256) void raster_kernel(
    const float* __restrict__ params_sorted,  // NG*8 floats
    const float* __restrict__ bg,             // 2 floats
    float* __restrict__ out) {                // 2*NPIX floats
  __shared__ float smem[NG * 8];  // 64 KB: all Gaussian params resident

  const int tid = threadIdx.x;
  const float TWO_PI = 6.28318530717958647692f;
  const float PI_F   = 3.14159265358979323846f;

#if USE_TDM
  // ---- TDM: one tensor_load_to_lds DMA stages the whole 64KB table ----
  if (tid == 0) {  // waves with EXEC==0 branch over this block
    unsigned lds_base = (unsigned)(unsigned long long)(uintptr_t)&smem[0];
    unsigned long long ga = (unsigned long long)(uintptr_t)params_sorted;
    u32x4 g0;
    g0.x = 1u;                                   // count=1 (valid), user mode
    g0.y = lds_base;                             // lds_addr
    g0.z = (unsigned)(ga & 0xffffffffull);       // global_addr[31:0]
    g0.w = (unsigned)((ga >> 32) & 0x1ffffffull) // global_addr[56:32]
           | (2u << 30);                         // type=2 ("image")
    i32x8 g1;
    g1[0] = 0x20000;                 // wg_mask=0, data_size=2 (4 bytes)
    g1[1] = (int)(16384u << 16);     // tensor_dim0[15:0]=16384
    g1[2] = (int)(1u << 16);         // tensor_dim0 hi=0 | tensor_dim1=1
    g1[3] = (int)(16384u << 16);     // tensor_dim1 hi=0 | tile_dim0=16384
    g1[4] = 1;                       // tile_dim1=1, tile_dim2=0
    g1[5] = 16384;                   // tensor_dim0_stride[31:0]
    g1[6] = (int)(16384u << 16);     // stride0 hi=0 | tensor_dim1_stride lo
    g1[7] = 0;                       // tensor_dim1_stride hi
    i32x4 z4 = {0, 0, 0, 0};
#if __clang_major__ >= 23
    i32x8 z8 = {0, 0, 0, 0, 0, 0, 0, 0};
    __builtin_amdgcn_tensor_load_to_lds(g0, g1, z4, z4, z8, 0);
#else
    __builtin_amdgcn_tensor_load_to_lds(g0, g1, z4, z4, 0);
#endif
    __builtin_amdgcn_s_wait_tensorcnt((short)0);
  }
  __syncthreads();
#else
  // ---- fallback: cooperative float4 staging ----
  {
    const float4* src = (const float4*)params_sorted;
    float4* dst = (float4*)smem;
    __builtin_prefetch(src + tid, 0, 0);  // global_prefetch_b8
    for (int k = tid; k < NG * 2; k += 256) dst[k] = src[k];
  }
  __syncthreads();
#endif

  const int lane = tid & 31;
  const int wave = tid >> 5;
  const int half = lane >> 4;  // which Gaussian pair of the group of 4
  const int col  = lane & 15;  // pixel row within the wave's 16 pixels

  const int pix_base = blockIdx.x * 128 + wave * 16;
  const int pixel = pix_base + col;
  const float theta_pix = ((float)(pixel / W_DIM) + 0.5f) * (PI_F / H_DIM);
  const float phi_pix =
      ((float)(pixel % W_DIM) + 0.5f) * (TWO_PI / W_DIM) - PI_F;

  // Branch-free B-matrix build: columns >=2 are zeroed by mask multiply.
  const float selmask = (col < 2) ? 1.0f : 0.0f;
  const int cidx = 6 + (col & 1);

  float T = 1.0f;
  v8f acc = {0.f, 0.f, 0.f, 0.f, 0.f, 0.f, 0.f, 0.f};

  for (int g = 0; g < NG; g += 4) {
    const int klo = g + half * 2;
    const float* q0 = &smem[klo * 8];
    const float* q1 = q0 + 8;

    float dt0 = theta_pix - q0[0];
    float dp0 = wrap_pi(phi_pix - q0[1]);
    float e0 = q0[2] * dt0 * dt0 + q0[3] * dt0 * dp0 + q0[4] * dp0 * dp0;
    float a_lo = fminf(fmaxf(q0[5] * __expf(e0), 0.0f), 0.99f);

    float dt1 = theta_pix - q1[0];
    float dp1 = wrap_pi(phi_pix - q1[1]);
    float e1 = q1[2] * dt1 * dt1 + q1[3] * dt1 * dp1 + q1[4] * dp1 * dp1;
    float a_hi = fminf(fmaxf(q1[5] * __expf(e1), 0.0f), 0.99f);

    // split transmittance scan across lane-halves (SWAPX16 swizzle)
    float om_lo = 1.0f - a_lo;
    float p = om_lo * (1.0f - a_hi);
    float p_other = __int_as_float(
        __builtin_amdgcn_ds_swizzle(__float_as_int(p), 0x401f));
    float prefix = half ? p_other : 1.0f;

    float w_lo = T * prefix * a_lo;
    float w_hi = T * prefix * om_lo * a_hi;
    T = T * p * p_other;

    v2f A;
    A.x = w_lo;  // K = 2*half
    A.y = w_hi;  // K = 2*half + 1
    v2f B;
    B.x = q0[cidx] * selmask;  // sig[klo][col]   (cols>=2 -> 0)
    B.y = q1[cidx] * selmask;  // sig[klo+1][col]

    acc = __builtin_amdgcn_wmma_f32_16x16x4_f32(
        false, A, false, B, (short)0, acc, false, false);
  }
  __syncthreads();

  // Extract C/D layout (VGPR r: lanes 0-15 -> M=r, lanes 16-31 -> M=8+r)
  float* ex = smem;  // reuse staging LDS
#pragma unroll
  for (int r = 0; r < 8; ++r) ex[tid * 8 + r] = acc[r];
  __syncthreads();

  {
    const int m = col;   // pixel row 0..15 in this wave
    const int n = half;  // signal component 0/1
    const int src_lane = wave * 32 + ((m < 8) ? n : (16 + n));
    float val = ex[src_lane * 8 + (m & 7)];
    val += T * bg[n];  // T identical across halves -> valid for pixel m
    out[n * NPIX + pix_base + m] = val;
  }
}

// ---------------------------------------------------------------------------
extern "C" void kernel_launch(void* const* d_in, const int* in_sizes, int n_in,
                              void* d_out, int out_size, void* d_ws,
                              size_t ws_size, hipStream_t stream) {
  const float* means = (const float*)d_in[0];
  const float* cov6  = (const float*)d_in[1];
  const float* sig   = (const float*)d_in[2];
  const float* att   = (const float*)d_in[3];
  const float* rad   = (const float*)d_in[4];
  const float* rx    = (const float*)d_in[5];
  const float* bg    = (const float*)d_in[6];
  float* out = (float*)d_out;

  float* dist   = (float*)d_ws;        // NG floats
  float* params = dist + NG;           // NG*8 floats
  float* sorted = params + NG * 8;     // NG*8 floats

  preprocess_kernel<<<NG / 256, 256, 0, stream>>>(means, cov6, sig, att, rad,
                                                  rx, dist, params);
  sort_scatter_kernel<<<NG / 256, 256, 0, stream>>>(dist, params, sorted);
  raster_kernel<<<NPIX / 128, 256, 0, stream>>>(sorted, bg, out);
}